// WholeRotationShiftedLoss_5231270167090
// MI455X (gfx1250) — compile-verified
//
#include <hip/hip_runtime.h>
#include <hip/hip_bf16.h>
#include <math.h>
#include <stdint.h>

// ---------------------------------------------------------------------------
// WholeRotationShiftedLoss for MI455X (gfx1250, wave32)
//   grid  = BS (2048) blocks, one batch image per block
//   block = 256 threads (8 wave32)
//   * TDM (tensor_load_to_lds) stages the two 64x64 f32 tiles into LDS once,
//     then 125 (shift,theta) configs re-use them (gather-bound inner loop).
//   * Per-wave reductions of (se, mk) partials via V_WMMA_F32_16X16X4_F32
//     ones-matrix trick (layout-safe: only uses documented A and C/D layouts).
// ---------------------------------------------------------------------------

typedef float v2f  __attribute__((ext_vector_type(2)));
typedef float v8f  __attribute__((ext_vector_type(8)));
typedef unsigned int u32x4 __attribute__((ext_vector_type(4)));
typedef int   i32x4 __attribute__((ext_vector_type(4)));
typedef int   i32x8 __attribute__((ext_vector_type(8)));

#define TILE   64
#define NPIX   (TILE * TILE)
#define BLOCK  256
#define SHIFT  2
#define NANGLE 2

__device__ __forceinline__ int iclamp(int v, int lo, int hi) {
  return v < lo ? lo : (v > hi ? hi : v);
}

// ---- TDM: DMA one 64x64 f32 tile (row-major, stride 64) global -> LDS ------
__device__ __forceinline__ void tdm_load_tile_64x64_f32(const float* gsrc,
                                                        unsigned lds_byte_off) {
  unsigned long long ga = (unsigned long long)(uintptr_t)gsrc;
  u32x4 g0;
  g0[0] = 1u;                                           // count=1 (valid user D#)
  g0[1] = lds_byte_off;                                 // lds_addr
  g0[2] = (unsigned)(ga & 0xFFFFFFFFull);               // global_addr[31:0]
  g0[3] = (unsigned)((ga >> 32) & 0x01FFFFFFull)        // global_addr[56:32]
        | (2u << 30);                                   // type = 2 ("image")
  i32x8 g1;
  g1[0] = (int)(2u << 16);       // data_size=2 (4B); no multicast/pad/iterate
  g1[1] = (int)(64u << 16);      // tensor_dim0[15:0]=64 (bits 63:48)
  g1[2] = (int)(64u << 16);      // tensor_dim0 hi=0; tensor_dim1[15:0]=64
  g1[3] = (int)(64u << 16);      // tensor_dim1 hi=0; tile_dim0=64
  g1[4] = 64;                    // tile_dim1=64; tile_dim2=0 (2D tile)
  g1[5] = 64;                    // tensor_dim0_stride=64 (low 32)
  g1[6] = (int)(4096u << 16);    // stride0 hi=0; tensor_dim1_stride[15:0]
  g1[7] = 0;                     // tensor_dim1_stride hi
  i32x4 z4 = {0, 0, 0, 0};
#if defined(__clang_major__) && (__clang_major__ >= 23)
  i32x8 z8 = {0, 0, 0, 0, 0, 0, 0, 0};
  __builtin_amdgcn_tensor_load_to_lds(g0, g1, z4, z4, z8, 0);
#else
  __builtin_amdgcn_tensor_load_to_lds(g0, g1, z4, z4, 0);
#endif
}

// ---- wave32 sum of one f32/lane via 5x V_WMMA_F32_16X16X4_F32 --------------
// Step 1: A=[x_m, 0 | x_{m+16}, 0], B=ones  -> D[m,n] = x_m + x_{m+16}
// Steps 2-5: A = (d[2j], d[2j+1]) pairs, B=ones, C accumulates
//            -> acc[*] = sum of all 32 lane values, broadcast to every lane.
__device__ __forceinline__ float wave_sum_f32_wmma(float x) {
  v2f ones; ones[0] = 1.0f; ones[1] = 1.0f;
  v2f a;    a[0] = x;       a[1] = 0.0f;
  v8f z = {};
  v8f d = __builtin_amdgcn_wmma_f32_16x16x4_f32(false, a, false, ones,
                                                (short)0, z, false, false);
  v8f acc = {};
  v2f p;
  p[0] = d[0]; p[1] = d[1];
  acc = __builtin_amdgcn_wmma_f32_16x16x4_f32(false, p, false, ones,
                                              (short)0, acc, false, false);
  p[0] = d[2]; p[1] = d[3];
  acc = __builtin_amdgcn_wmma_f32_16x16x4_f32(false, p, false, ones,
                                              (short)0, acc, false, false);
  p[0] = d[4]; p[1] = d[5];
  acc = __builtin_amdgcn_wmma_f32_16x16x4_f32(false, p, false, ones,
                                              (short)0, acc, false, false);
  p[0] = d[6]; p[1] = d[7];
  acc = __builtin_amdgcn_wmma_f32_16x16x4_f32(false, p, false, ones,
                                              (short)0, acc, false, false);
  return acc[0];
}

__global__ __launch_bounds__(BLOCK)
void WholeRotationShiftedLoss_5231270167090_kernel(const float* __restrict__ fm1,
                                                   const float* __restrict__ fm2,
                                                   float* __restrict__ out) {
  __shared__ float s1[NPIX];
  __shared__ float s2[NPIX];
  __shared__ float sWse[8];
  __shared__ float sWmk[8];
  __shared__ float sMin;

  const int b   = blockIdx.x;
  const int tid = threadIdx.x;

  // Stage both tiles into LDS with the Tensor Data Mover (wave 0 issues).
  if (tid < 32) {
    unsigned off1 = (unsigned)(uintptr_t)(void*)&s1[0];  // flat low32 = LDS offset
    unsigned off2 = (unsigned)(uintptr_t)(void*)&s2[0];
    tdm_load_tile_64x64_f32(fm1 + (size_t)b * NPIX, off1);
    tdm_load_tile_64x64_f32(fm2 + (size_t)b * NPIX, off2);
    __builtin_amdgcn_s_wait_tensorcnt(0);
  }
  if (tid == 0) sMin = 3.402823466e+38f;
  __syncthreads();

  const float DEG = 0.017453292519943295f;  // pi/180

  for (int bh = -SHIFT; bh <= SHIFT; ++bh) {
    const int ow    = TILE - (bh < 0 ? -bh : bh);
    const int x1off = bh < 0 ? -bh : 0;
    const int x2off = bh >= 0 ? bh : 0;
    for (int bv = -SHIFT; bv <= SHIFT; ++bv) {
      const int oh    = TILE - (bv < 0 ? -bv : bv);
      const int y1off = bv < 0 ? -bv : 0;
      const int y2off = bv >= 0 ? bv : 0;
      const int npix  = oh * ow;
      const int jstep = BLOCK - 4 * ow;        // 256 = 4*ow + jstep, jstep in [0,16]
      const float cx = 0.5f * (float)ow;
      const float cy = 0.5f * (float)oh;
      for (int th = -NANGLE; th <= NANGLE; ++th) {
        const float t     = -(float)th * DEG;  // cv2 inverse map: rotate by -theta
        const float alpha = __cosf(t);
        const float beta  = __sinf(t);
        const float cxa = (1.0f - alpha) * cx - beta * cy;
        const float cya = beta * cx + (1.0f - alpha) * cy;

        float se = 0.0f, mk = 0.0f;
        int i = 0, j = tid;
        while (j >= ow) { j -= ow; ++i; }      // <= 4 iterations
        for (int p = tid; p < npix; p += BLOCK) {
          const float jf = (float)j, iff = (float)i;
          const float xs = alpha * jf + beta * iff + cxa;
          const float ys = alpha * iff - beta * jf + cya;
          const float x0f = floorf(xs);
          const float y0f = floorf(ys);
          const float wx = xs - x0f;
          const float wy = ys - y0f;
          const int x0 = (int)x0f, y0 = (int)y0f;
          const int x1 = x0 + 1,  y1 = y0 + 1;
          const float vx0 = ((unsigned)x0 < (unsigned)ow) ? 1.0f : 0.0f;
          const float vx1 = ((unsigned)x1 < (unsigned)ow) ? 1.0f : 0.0f;
          const float vy0 = ((unsigned)y0 < (unsigned)oh) ? 1.0f : 0.0f;
          const float vy1 = ((unsigned)y1 < (unsigned)oh) ? 1.0f : 0.0f;
          const int xc0 = iclamp(x0, 0, ow - 1) + x2off;
          const int xc1 = iclamp(x1, 0, ow - 1) + x2off;
          const int yr0 = (iclamp(y0, 0, oh - 1) + y2off) * TILE;
          const int yr1 = (iclamp(y1, 0, oh - 1) + y2off) * TILE;
          const float w00 = (1.0f - wy) * (1.0f - wx) * (vy0 * vx0);
          const float w01 = (1.0f - wy) * wx          * (vy0 * vx1);
          const float w10 = wy          * (1.0f - wx) * (vy1 * vx0);
          const float w11 = wy          * wx          * (vy1 * vx1);
          float acc = w00 * s2[yr0 + xc0];
          acc = fmaf(w01, s2[yr0 + xc1], acc);
          acc = fmaf(w10, s2[yr1 + xc0], acc);
          acc = fmaf(w11, s2[yr1 + xc1], acc);
          const float mloc = (w00 + w01) + (w10 + w11);
          const float r1v  = s1[(i + y1off) * TILE + (j + x1off)];
          const float dlt  = r1v - acc;
          se = fmaf(dlt * dlt, mloc, se);
          mk += mloc;
          // advance (i,j) by 256 pixels row-major
          i += 4; j += jstep;
          if (j >= ow) { j -= ow; ++i; }
        }

        // wave-level WMMA reduction (full EXEC here), then cross-wave via LDS
        const float wse = wave_sum_f32_wmma(se);
        const float wmk = wave_sum_f32_wmma(mk);
        const int wave = tid >> 5, lane = tid & 31;
        if (lane == 0) { sWse[wave] = wse; sWmk[wave] = wmk; }
        __syncthreads();
        if (tid == 0) {
          float ts = 0.0f, tm = 0.0f;
#pragma unroll
          for (int w = 0; w < 8; ++w) { ts += sWse[w]; tm += sWmk[w]; }
          sMin = fminf(sMin, ts / tm);
        }
        __syncthreads();
      }
    }
  }
  if (tid == 0) out[b] = sMin;
}

extern "C" void kernel_launch(void* const* d_in, const int* in_sizes, int n_in,
                              void* d_out, int out_size, void* d_ws, size_t ws_size,
                              hipStream_t stream) {
  (void)in_sizes; (void)n_in; (void)d_ws; (void)ws_size;
  const float* fm1 = (const float*)d_in[0];
  const float* fm2 = (const float*)d_in[1];
  float* out = (float*)d_out;
  const int bs = out_size;  // 2048
  WholeRotationShiftedLoss_5231270167090_kernel<<<dim3(bs), dim3(BLOCK), 0, stream>>>(
      fm1, fm2, out);
}